// SparseConv2DTranspose_70222715290211
// MI455X (gfx1250) — compile-verified
//
#include <hip/hip_runtime.h>

// ---------------------------------------------------------------------------
// SparseConv2DTranspose, KS=2 ST=2 => non-overlapping deconv:
//   out[n,2i+di,2j+dj,f] = act(block) * (bias[f] + sum_c in[n,i,j,c]*K[di,dj,f,c])
// Four 64x64 fp32 GEMMs over 524288 pixels via V_WMMA_F32_16X16X4_F32.
// Memory-bound (~673MB @ 23.3TB/s ~= 29us); fp32 WMMA keeps compute at the
// same roofline, so no precision downgrade is warranted.
// ---------------------------------------------------------------------------

typedef float v2f __attribute__((ext_vector_type(2)));
typedef float v8f __attribute__((ext_vector_type(8)));
typedef float f32x4 __attribute__((ext_vector_type(4)));
typedef int   v4i __attribute__((ext_vector_type(4)));

// Pointer-to-int4 in global (AS1) / LDS (AS3) address spaces, matching the
// builtin's parameter types reported by hipcc.
typedef __attribute__((address_space(1))) v4i* v4i_gp;
typedef __attribute__((address_space(3))) v4i* v4i_lp;

#define N_  8
#define H_  256
#define W_  256
#define C_  64
#define F_  64
#define BS_ 16
#define NBLK (N_ * (H_ / BS_) * (W_ / BS_))   // 8*16*16 = 2048

// ---------------------------------------------------------------------------
// Kernel 1: per 16x16 block max-reduce of the mask, threshold 0.5 -> flag
// ---------------------------------------------------------------------------
__global__ __launch_bounds__(256)
void mask_reduce_kernel(const float* __restrict__ mask, float* __restrict__ act) {
    __shared__ float red[256];
    int b  = blockIdx.x;                 // 0..2047
    int n  = b >> 8;
    int bh = (b >> 4) & 15;
    int bw = b & 15;
    int t  = threadIdx.x;                // 0..255
    int r  = t >> 4, c = t & 15;
    red[t] = mask[(size_t)((n * H_ + bh * BS_ + r) * W_ + bw * BS_ + c)];
    __syncthreads();
    for (int s = 128; s > 0; s >>= 1) {
        if (t < s) red[t] = fmaxf(red[t], red[t + s]);
        __syncthreads();
    }
    if (t == 0) act[b] = (red[0] > 0.5f) ? 1.0f : 0.0f;
}

// ---------------------------------------------------------------------------
// Kernel 2: one workgroup (16 waves) per 16x16 input block.
//  - stage 64KB of weights (4 taps x 64F x 64C fp32) into LDS (async path)
//  - wave w handles block row r=w: 16 pixels -> A 16x64 fp32 in 32 VGPRs
//  - per (tap, ftile): 16x v_wmma_f32_16x16x4_f32, +bias, *act, store C tile
// ---------------------------------------------------------------------------
__global__ __launch_bounds__(512)
void deconv_wmma_kernel(const float* __restrict__ in,
                        const float* __restrict__ wgt,   // [2,2,F,C]
                        const float* __restrict__ bias,  // [F]
                        const float* __restrict__ act,   // [2048]
                        float* __restrict__ out) {       // [N,512,512,F]
    __shared__ float lw[4 * F_ * C_];    // 64 KB

    int b  = blockIdx.x;
    int n  = b >> 8;
    int bh = (b >> 4) & 15;
    int bw = b & 15;
    int t  = threadIdx.x;

    // ---- stage weights into LDS: 4096 float4 units, 512 threads x 8 ----
#if __has_builtin(__builtin_amdgcn_global_load_async_to_lds_b128)
    #pragma unroll
    for (int k = 0; k < 8; ++k) {
        int u = k * 512 + t;             // float4 unit index
        __builtin_amdgcn_global_load_async_to_lds_b128(
            (v4i_gp)(wgt + (size_t)u * 4),
            (v4i_lp)(lw + (size_t)u * 4),
            0, 0);
    }
  #if __has_builtin(__builtin_amdgcn_s_wait_asynccnt)
    __builtin_amdgcn_s_wait_asynccnt(0);
  #else
    asm volatile("s_wait_asynccnt 0" ::: "memory");
  #endif
#else
    #pragma unroll
    for (int k = 0; k < 8; ++k) {
        int u = k * 512 + t;
        ((f32x4*)lw)[u] = ((const f32x4*)wgt)[u];
    }
#endif
    __syncthreads();

    int lane = t & 31;
    int wv   = t >> 5;                   // wave id == block row r (0..15)
    int p    = lane & 15;                // pixel within row / column of tile
    int koff = (lane >> 4) << 1;         // 0 for lanes 0-15, 2 for lanes 16-31

    // ---- A fragments: 16 pixels x 64 channels, ISA 32-bit A layout:
    //      slice s VGPR0 = K=4s (lanes 0-15) / K=4s+2 (lanes 16-31), VGPR1 = +1
    const float* inp =
        in + (size_t)((n * H_ + bh * BS_ + wv) * W_ + bw * BS_ + p) * C_;
    v2f a[16];
    #pragma unroll
    for (int s = 0; s < 16; ++s)
        a[s] = *(const v2f*)(inp + 4 * s + koff);

    float av = act[b];
    int   i  = bh * BS_ + wv;            // input row
    int   jb = bw * BS_;                 // input col base

    #pragma unroll
    for (int tap = 0; tap < 4; ++tap) {
        int di = tap >> 1, dj = tap & 1;
        size_t orow = ((size_t)(n * 2 * H_) + 2 * i + di) * (size_t)(2 * W_);

        #pragma unroll
        for (int ft = 0; ft < 4; ++ft) {
            int f = ft * 16 + p;         // output channel this lane owns (N dim)
            // B fragment base in LDS: row k=c, col n=f; B VGPR0 = {K0|K2} halves
            const float* lwb = lw + ((tap * F_ + f) * C_) + koff;

            v8f cacc = {};
            #pragma unroll
            for (int s = 0; s < 16; ++s) {
                v2f bfrag = *(const v2f*)(lwb + 4 * s);   // ds_load_b64
                cacc = __builtin_amdgcn_wmma_f32_16x16x4_f32(
                    /*neg_a=*/false, a[s], /*neg_b=*/false, bfrag,
                    /*c_mod=*/(short)0, cacc,
                    /*reuse_a=*/false, /*reuse_b=*/false);
            }

            float bb = bias[f];
            // C layout: element e -> pixel M = e + (lane>=16 ? 8 : 0), N = lane%16
            #pragma unroll
            for (int e = 0; e < 8; ++e) {
                int pp = e + ((lane >> 4) << 3);
                size_t oidx = (orow + (size_t)(2 * (jb + pp) + dj)) * F_ + f;
                out[oidx] = (cacc[e] + bb) * av;
            }
        }
    }
}

// ---------------------------------------------------------------------------
extern "C" void kernel_launch(void* const* d_in, const int* in_sizes, int n_in,
                              void* d_out, int out_size, void* d_ws, size_t ws_size,
                              hipStream_t stream) {
    (void)in_sizes; (void)n_in; (void)out_size; (void)ws_size;
    const float* inputs = (const float*)d_in[0];   // [8,256,256,64]
    const float* mask   = (const float*)d_in[1];   // [8,256,256,1]
    const float* kern   = (const float*)d_in[2];   // [2,2,64,64]
    const float* bias   = (const float*)d_in[3];   // [64]
    float*       out    = (float*)d_out;           // [8,512,512,64]
    float*       actws  = (float*)d_ws;            // 2048 flags

    mask_reduce_kernel<<<NBLK, 256, 0, stream>>>(mask, actws);
    deconv_wmma_kernel<<<NBLK, 512, 0, stream>>>(inputs, kern, bias, actws, out);
}